// MergeHead_62517543960956
// MI455X (gfx1250) — compile-verified
//
#include <hip/hip_runtime.h>
#include <math.h>

typedef __attribute__((ext_vector_type(2))) float v2f;
typedef __attribute__((ext_vector_type(4))) float v4f;
typedef __attribute__((ext_vector_type(8))) float v8f;

#define NEG_INF_F (-100000000.0f)

static __device__ __forceinline__ v8f wmma_f32(v2f a, v2f b, v8f c) {
    // V_WMMA_F32_16X16X4_F32 : D = A(16x4) * B(4x16) + C(16x16), all f32
    return __builtin_amdgcn_wmma_f32_16x16x4_f32(
        false, a, false, b, (short)0, c, false, false);
}

static __device__ __forceinline__ v2f ld2(const float* p) {
    return *(const v2f*)p;   // 8B-aligned by construction -> global_load_b64
}

// ---------------------------------------------------------------------------
// LDS-tiled transpose: feats [b][c][p] (256 x 4096) -> feats_t [b][p][c]
// grid (128 p-tiles, 8 c-tiles, B), block (32, 8)
// ---------------------------------------------------------------------------
__global__ void __launch_bounds__(256)
transpose_feats_k(const float* __restrict__ in, float* __restrict__ out)
{
    __shared__ float tile[32][33];
    const int b  = blockIdx.z;
    const int pT = blockIdx.x << 5;
    const int cT = blockIdx.y << 5;
    const int tx = threadIdx.x, ty = threadIdx.y;
    const float* ib = in  + (size_t)b * 256 * 4096;
    float*       ob = out + (size_t)b * 4096 * 256;
    #pragma unroll
    for (int j = 0; j < 32; j += 8)
        tile[ty + j][tx] = ib[(size_t)(cT + ty + j) * 4096 + pT + tx];
    __syncthreads();
    #pragma unroll
    for (int j = 0; j < 32; j += 8)
        ob[(size_t)(pT + ty + j) * 256 + cT + tx] = tile[tx][ty + j];
}

// w [co][ci][ky][kx] -> wt [tap][co][ci]   (9*256*256 elements)
__global__ void __launch_bounds__(256)
transpose_w3_k(const float* __restrict__ w, float* __restrict__ wt)
{
    const int idx = blockIdx.x * 256 + threadIdx.x;
    const int ci  = idx & 255;
    const int co  = (idx >> 8) & 255;
    const int tap = idx >> 16;
    const int ky = tap / 3, kx = tap % 3;
    wt[idx] = w[(((size_t)co * 256 + ci) * 3 + ky) * 3 + kx];
}

// ---------------------------------------------------------------------------
// 3x3 conv (pad=1) + bias + ReLU, activations in [b][p][c] layout.
// One wave per 32(ch) x 16(pixel) tile (2 co-tiles, B reused).
// grid = (B*256 strips, 8), block = 32.
// ---------------------------------------------------------------------------
__global__ void __launch_bounds__(32)
conv3x3_relu_k(const float* __restrict__ in_t, const float* __restrict__ wt,
               const float* __restrict__ bias, float* __restrict__ out_t)
{
    const int lane  = threadIdx.x;
    const int half  = lane >> 4;
    const int idx16 = lane & 15;
    const int kbase = half * 2;

    const int bIdx = blockIdx.x >> 8;
    const int p0   = (blockIdx.x & 255) << 4;
    const int co0  = blockIdx.y << 5;          // 2 consecutive co-tiles

    const int y = p0 >> 6;                     // wave-uniform row
    const int x = (p0 & 63) + idx16;

    const float* inb = in_t + (size_t)bIdx * 4096 * 256;

    v8f acc0 = {}, acc1 = {};
    for (int ky = 0; ky < 3; ++ky) {
        const int ys = y + ky - 1;
        if (ys < 0 || ys > 63) continue;       // uniform: no divergence
        for (int kx = 0; kx < 3; ++kx) {
            const int  xs = x + kx - 1;
            const float vf = (xs >= 0 && xs < 64) ? 1.0f : 0.0f;
            const int  xc = xs < 0 ? 0 : (xs > 63 ? 63 : xs);
            const float* bptr  = inb + (size_t)(ys * 64 + xc) * 256 + kbase;
            const float* aptr0 = wt + ((size_t)((ky * 3 + kx) * 256) + co0 + idx16) * 256 + kbase;
            const float* aptr1 = aptr0 + 16 * 256;
            for (int ci = 0; ci < 256; ci += 4) {
                v2f bb = ld2(bptr + ci);
                bb[0] *= vf;
                bb[1] *= vf;
                v2f a0 = ld2(aptr0 + ci);
                v2f a1 = ld2(aptr1 + ci);
                acc0 = wmma_f32(a0, bb, acc0);
                acc1 = wmma_f32(a1, bb, acc1);
            }
        }
    }

    float* ob = out_t + ((size_t)bIdx * 4096 + p0 + idx16) * 256;
    #pragma unroll
    for (int t = 0; t < 2; ++t) {
        const v8f& A = t ? acc1 : acc0;
        const int c0 = co0 + t * 16 + half * 8;
        v4f s0, s1;
        #pragma unroll
        for (int r = 0; r < 4; ++r) {
            float v0 = A[r] + bias[c0 + r];
            float v1 = A[4 + r] + bias[c0 + 4 + r];
            s0[r] = v0 > 0.0f ? v0 : 0.0f;
            s1[r] = v1 > 0.0f ? v1 : 0.0f;
        }
        *(v4f*)(ob + c0)     = s0;
        *(v4f*)(ob + c0 + 4) = s1;
    }
}

// ---------------------------------------------------------------------------
// 1x1 conv + bias, in_t [b][p][c] -> out_t [b][p][ostride] (Cout = 256/257)
// grid = (B*256 strips, ceil(Cout/16)), block = 32.
// ---------------------------------------------------------------------------
__global__ void __launch_bounds__(32)
conv1x1_k(const float* __restrict__ in_t, const float* __restrict__ w,
          const float* __restrict__ bias, float* __restrict__ out_t,
          int Cout, int ostride)
{
    const int lane  = threadIdx.x;
    const int half  = lane >> 4;
    const int idx16 = lane & 15;
    const int kbase = half * 2;

    const int bIdx = blockIdx.x >> 8;
    const int p0   = (blockIdx.x & 255) << 4;
    const int co0  = blockIdx.y << 4;

    const int  row = co0 + idx16;
    const bool rowValid = row < Cout;
    const float avf = rowValid ? 1.0f : 0.0f;
    const float* aptr = w + (size_t)(rowValid ? row : 0) * 256 + kbase;
    const float* bptr = in_t + ((size_t)bIdx * 4096 + p0 + idx16) * 256 + kbase;

    v8f acc = {};
    for (int ci = 0; ci < 256; ci += 4) {
        v2f a = ld2(aptr + ci);
        a[0] *= avf;
        a[1] *= avf;
        v2f bb = ld2(bptr + ci);
        acc = wmma_f32(a, bb, acc);
    }

    float* ob = out_t + ((size_t)bIdx * 4096 + p0 + idx16) * (size_t)ostride;
    const int c0 = co0 + half * 8;
    if (co0 + 16 <= Cout) {
        v4f s0, s1;
        #pragma unroll
        for (int r = 0; r < 4; ++r) {
            s0[r] = acc[r]     + bias[c0 + r];
            s1[r] = acc[4 + r] + bias[c0 + 4 + r];
        }
        *(v4f*)(ob + c0)     = s0;
        *(v4f*)(ob + c0 + 4) = s1;
    } else {
        #pragma unroll
        for (int r = 0; r < 8; ++r) {
            const int cm = c0 + r;
            if (cm < Cout) ob[cm] = acc[r] + bias[cm];
        }
    }
}

// ---------------------------------------------------------------------------
// logits[b,p,q] = sum_c kert[b,p,c] * ft[b,q,c] + kert[b,p,256]
// One wave per 16(p) x 64(q) (4 q-tiles, A reused). Fused dice-loss partials.
// grid = (64, 256, B), block = 32.
// ---------------------------------------------------------------------------
__global__ void __launch_bounds__(32)
logits_loss_k(const float* __restrict__ kert, const float* __restrict__ ft,
              const float* __restrict__ masks, const int* __restrict__ layouts,
              float* __restrict__ out, float* __restrict__ num_acc)
{
    const int lane  = threadIdx.x;
    const int half  = lane >> 4;
    const int idx16 = lane & 15;
    const int kbase = half * 2;

    const int q00 = blockIdx.x << 6;
    const int p0  = blockIdx.y << 4;
    const int b   = blockIdx.z;

    const float* kb_base = kert + (size_t)b * 4096 * 260;
    const float* kA = kb_base + (size_t)(p0 + idx16) * 260 + kbase;
    const float* fB = ft + ((size_t)b * 4096 + q00 + idx16) * 256 + kbase;

    v8f acc0 = {}, acc1 = {}, acc2 = {}, acc3 = {};
    for (int ci = 0; ci < 256; ci += 4) {
        v2f a  = ld2(kA + ci);
        v2f b0 = ld2(fB + ci);
        v2f b1 = ld2(fB + 16 * 256 + ci);
        v2f b2 = ld2(fB + 32 * 256 + ci);
        v2f b3 = ld2(fB + 48 * 256 + ci);
        acc0 = wmma_f32(a, b0, acc0);
        acc1 = wmma_f32(a, b1, acc1);
        acc2 = wmma_f32(a, b2, acc2);
        acc3 = wmma_f32(a, b3, acc3);
    }

    // per-p-row constants (shared by all 4 q-tiles)
    float kbr[8], mp[8];
    int   lp[8];
    #pragma unroll
    for (int r = 0; r < 8; ++r) {
        const int p = p0 + r + half * 8;
        kbr[r] = kb_base[(size_t)p * 260 + 256];
        mp[r]  = masks[b * 4096 + p];
        lp[r]  = layouts[b * 4096 + p];
    }

    float wsum = 0.0f;
    #pragma unroll
    for (int t = 0; t < 4; ++t) {
        const v8f& A = (t == 0) ? acc0 : (t == 1) ? acc1 : (t == 2) ? acc2 : acc3;
        const int   q  = q00 + t * 16 + idx16;
        const float mq = masks[b * 4096 + q];
        const int   lq = layouts[b * 4096 + q];
        #pragma unroll
        for (int r = 0; r < 8; ++r) {
            const int   p     = p0 + r + half * 8;
            const float logit = A[r] + kbr[r];
            out[((size_t)(b * 4096 + p)) * 4096 + q] = (mq > 0.0f) ? logit : NEG_INF_F;
            // branch-free dice term; zeroed by mq when mask==0
            const float lab = (lp[r] == lq) ? 1.0f : 0.0f;
            const float ps  = 1.0f / (1.0f + __expf(-logit));
            const float dd  = 2.0f * ps * lab / (ps * ps + lab + 0.002f);
            wsum += (1.0f - dd) * mq * mp[r];
        }
    }

    for (int off = 16; off > 0; off >>= 1)
        wsum += __shfl_xor(wsum, off);
    if (lane == 0)
        atomicAdd(&num_acc[b], wsum);
}

__global__ void init_acc_k(float* acc) {
    if (threadIdx.x < 2) acc[threadIdx.x] = 0.0f;
}

__global__ void __launch_bounds__(256)
finalize_k(const float* __restrict__ masks, const float* __restrict__ num_acc,
           float* __restrict__ out_loss)
{
    __shared__ float r0[256];
    __shared__ float r1[256];
    float s0 = 0.0f, s1 = 0.0f;
    for (int i = threadIdx.x; i < 4096; i += 256) {
        s0 += masks[i];
        s1 += masks[4096 + i];
    }
    r0[threadIdx.x] = s0;
    r1[threadIdx.x] = s1;
    __syncthreads();
    for (int s = 128; s > 0; s >>= 1) {
        if (threadIdx.x < s) {
            r0[threadIdx.x] += r0[threadIdx.x + s];
            r1[threadIdx.x] += r1[threadIdx.x + s];
        }
        __syncthreads();
    }
    if (threadIdx.x == 0) {
        const float g0 = r0[0], g1 = r1[0];
        out_loss[0] = 0.5f * (num_acc[0] / (g0 * g0 + 1e-5f) +
                              num_acc[1] / (g1 * g1 + 1e-5f));
    }
}

extern "C" void kernel_launch(void* const* d_in, const int* in_sizes, int n_in,
                              void* d_out, int out_size, void* d_ws, size_t ws_size,
                              hipStream_t stream)
{
    const float* feats   = (const float*)d_in[0];
    const float* masks   = (const float*)d_in[1];
    const int*   layouts = (const int*)  d_in[2];
    const float* w_pre0  = (const float*)d_in[3];
    const float* b_pre0  = (const float*)d_in[4];
    const float* w_pre1  = (const float*)d_in[5];
    const float* b_pre1  = (const float*)d_in[6];
    const float* w_ker   = (const float*)d_in[7];
    const float* b_ker   = (const float*)d_in[8];
    const float* w_f     = (const float*)d_in[9];
    const float* b_f     = (const float*)d_in[10];

    float* out = (float*)d_out;
    float* ws  = (float*)d_ws;

    // workspace layout (floats)
    float* featsT = ws;                  // 2*4096*256 = 2,097,152
    float* h0     = ws + 2097152;        // 2,097,152
    float* h1     = ws + 4194304;        // 2,097,152
    float* kerT   = ws + 6291456;        // 2*4096*260 = 2,129,920 (stride 260, [256]=bias row)
    float* fT     = ws + 8421376;        // 2,097,152
    float* wt0    = ws + 10518528;       // 9*256*256 = 589,824
    float* wt1    = ws + 11108352;       // 589,824
    float* acc    = ws + 11698176;       // 2

    init_acc_k<<<dim3(1), dim3(32), 0, stream>>>(acc);

    transpose_feats_k<<<dim3(128, 8, 2), dim3(32, 8), 0, stream>>>(feats, featsT);
    transpose_w3_k<<<dim3(2304), dim3(256), 0, stream>>>(w_pre0, wt0);
    transpose_w3_k<<<dim3(2304), dim3(256), 0, stream>>>(w_pre1, wt1);

    conv3x3_relu_k<<<dim3(512, 8), dim3(32), 0, stream>>>(featsT, wt0, b_pre0, h0);
    conv3x3_relu_k<<<dim3(512, 8), dim3(32), 0, stream>>>(h0, wt1, b_pre1, h1);

    conv1x1_k<<<dim3(512, 17), dim3(32), 0, stream>>>(h1, w_ker, b_ker, kerT, 257, 260);
    conv1x1_k<<<dim3(512, 16), dim3(32), 0, stream>>>(featsT, w_f, b_f, fT, 256, 256);

    logits_loss_k<<<dim3(64, 256, 2), dim3(32), 0, stream>>>(kerT, fT, masks, layouts,
                                                             out, acc);

    finalize_k<<<dim3(1), dim3(256), 0, stream>>>(masks, acc,
                                                  out + (size_t)2 * 4096 * 4096);
}